// ProjKmeans_11665131176019
// MI455X (gfx1250) — compile-verified
//
#include <hip/hip_runtime.h>

// ProjKmeans fused pipeline for MI455X (gfx1250, wave32, WMMA f32 16x16x4).
//
// Shapes: B=16384, D=512, M=8, K=256, Dout=64, T=0.1
// Outputs (concat, fp32): Xr[B,M,D], Xp[B,M,D], Xr_sum[B,D], Xp_sum[B,D],
//                         center[K,D], label[B,M] (stored as float)

typedef __attribute__((ext_vector_type(2))) float v2f;
typedef __attribute__((ext_vector_type(4))) float v4f;
typedef __attribute__((ext_vector_type(8))) float v8f;

#define B_TOT 16384
#define D_DIM 512
#define M_DIM 8
#define K_CNT 256
#define DOUT  64

#define XT_PITCH 516   // X tile / Xp-sum LDS row pitch (floats): bank-conflict free, 16B aligned
#define AT_PITCH 68    // a-tile LDS row pitch (floats): conflict-free b64 frag loads, 16B aligned

static __device__ __forceinline__ v8f wmma4(v2f a, v2f b, v8f c) {
  // D(16x16,f32) = A(16x4,f32) x B(4x16,f32) + C
  return __builtin_amdgcn_wmma_f32_16x16x4_f32(
      /*neg_a=*/false, a, /*neg_b=*/false, b,
      /*c_mod=*/(short)0, c, /*reuse_a=*/false, /*reuse_b=*/false);
}

// ---------------- prep 1: cproj[m][k][o] = sum_d W[m][d][o] * C[k][d] ----------------
__global__ __launch_bounds__(64) void prep_cproj(const float* __restrict__ W,
                                                 const float* __restrict__ C,
                                                 float* __restrict__ cproj) {
  const int blk = blockIdx.x;          // m*256 + k
  const int m = blk >> 8, k = blk & 255;
  const int o = threadIdx.x;           // 0..63
  const float* wm = W + (size_t)m * D_DIM * DOUT + o;
  const float* ck = C + (size_t)k * D_DIM;
  float acc = 0.f;
  #pragma unroll 8
  for (int d = 0; d < D_DIM; ++d) acc += wm[(size_t)d * DOUT] * ck[d];
  cproj[((size_t)m * K_CNT + k) * DOUT + o] = acc;
}

// ---------------- prep 2: cnorm[k] = ||C[k]||^2 ; copy center to output ----------------
__global__ __launch_bounds__(128) void prep_cnorm(const float* __restrict__ C,
                                                  float* __restrict__ cnorm,
                                                  float* __restrict__ out_center) {
  __shared__ float red[128];
  const int k = blockIdx.x, t = threadIdx.x;
  float s = 0.f;
  for (int d = t; d < D_DIM; d += 128) {
    float v = C[(size_t)k * D_DIM + d];
    out_center[(size_t)k * D_DIM + d] = v;
    s += v * v;
  }
  red[t] = s;
  __syncthreads();
  for (int st = 64; st > 0; st >>= 1) {
    if (t < st) red[t] += red[t + st];
    __syncthreads();
  }
  if (t == 0) cnorm[k] = red[0];
}

// ---------------- main: projection + kmeans assignment, 64-batch tile per WG ----------------
__global__ __launch_bounds__(128) void projkmeans_main(
    const float* __restrict__ X,      // [B,512]
    const float* __restrict__ W,      // [8,512,64]
    const float* __restrict__ cproj,  // [8,256,64]
    const float* __restrict__ cnorm,  // [256]
    const float* __restrict__ center, // [256,512]
    float* __restrict__ o_xr, float* __restrict__ o_xp,
    float* __restrict__ o_xrs, float* __restrict__ o_xps,
    float* __restrict__ o_lbl) {
  extern __shared__ float smem[];
  float* xt  = smem;                                    // [64][516]: X tile, later Xp-sum
  float* at  = smem + 64 * XT_PITCH;                    // [8][64][68]: a, b-major
  int*   lbl = (int*)(smem + 64 * XT_PITCH + M_DIM * 64 * AT_PITCH); // [8][64]
  float* lcn = (float*)(lbl + M_DIM * 64);              // [256]

  const int tid  = threadIdx.x;
  const int lane = tid & 31;
  const int wv   = tid >> 5;     // 0..3
  const int ln   = lane & 15;    // tile column / row lane
  const int half = lane >> 4;    // WMMA lane-half
  const int b0   = blockIdx.x * 64;
  const int wb   = wv * 16;      // wave's local batch-column base

  // ---- phase 1: stage X tile row-major (pitch 516) + cnorm into LDS ----
  {
    const int row = tid >> 1, hh = tid & 1;
    const float* src = X + (size_t)(b0 + row) * D_DIM + hh * 256;
    float* dst = xt + row * XT_PITCH + hh * 256;
    #pragma unroll 8
    for (int i = 0; i < 64; ++i)
      *(v4f*)(dst + i * 4) = *(const v4f*)(src + i * 4);
    lcn[tid]       = cnorm[tid];
    lcn[tid + 128] = cnorm[tid + 128];
  }
  __syncthreads();

  // ---- phase 2: a[m] (64x64) = W_m^T (64x512) * Xt (512x64); wave owns o-chunk ----
  const int o0 = wv * 16;
  for (int m = 0; m < M_DIM; ++m) {
    v8f acc[4] = {};
    const float* wbase = W + (size_t)m * D_DIM * DOUT;
    for (int kd = 0; kd < 128; ++kd) {
      const int dd = kd * 4 + 2 * half;       // K offsets {dd, dd+1} for this lane-half
      v2f af;
      af.x = wbase[(size_t)dd * DOUT + o0 + ln];         // A[o][d] = W[m][d][o]
      af.y = wbase[(size_t)(dd + 1) * DOUT + o0 + ln];
      v2f bfr0 = *(const v2f*)(xt + (0 * 16 + ln) * XT_PITCH + dd);
      v2f bfr1 = *(const v2f*)(xt + (1 * 16 + ln) * XT_PITCH + dd);
      v2f bfr2 = *(const v2f*)(xt + (2 * 16 + ln) * XT_PITCH + dd);
      v2f bfr3 = *(const v2f*)(xt + (3 * 16 + ln) * XT_PITCH + dd);
      acc[0] = wmma4(af, bfr0, acc[0]);
      acc[1] = wmma4(af, bfr1, acc[1]);
      acc[2] = wmma4(af, bfr2, acc[2]);
      acc[3] = wmma4(af, bfr3, acc[3]);
    }
    const int obase = o0 + half * 8;          // lane's 8 D-rows are contiguous in o
    #pragma unroll
    for (int c = 0; c < 4; ++c) {
      float* p = at + (m * 64 + c * 16 + ln) * AT_PITCH + obase;
      v4f lo = {acc[c][0], acc[c][1], acc[c][2], acc[c][3]};
      v4f hi = {acc[c][4], acc[c][5], acc[c][6], acc[c][7]};
      *(v4f*)p = lo;
      *(v4f*)(p + 4) = hi;
    }
  }
  __syncthreads();

  // ---- zero Xp-sum region (wave-private columns; X tile no longer needed) ----
  {
    v4f z = {};
    #pragma unroll 4
    for (int dc = 0; dc < 32; ++dc) {
      float* p = xt + (wb + ln) * XT_PITCH + dc * 16 + half * 8;
      *(v4f*)p = z;
      *(v4f*)(p + 4) = z;
    }
  }

  // ---- phases 3+4 per m: Xp GEMM + score GEMM/argmin; wave owns its 16 columns ----
  const int bcol = b0 + wb + ln;
  for (int m = 0; m < M_DIM; ++m) {
    // cache B-fragments of a (K=o dimension, 16 steps) in registers
    v2f bf[16];
    #pragma unroll
    for (int t = 0; t < 16; ++t)
      bf[t] = *(const v2f*)(at + (m * 64 + wb + ln) * AT_PITCH + t * 4 + 2 * half);

    const float* wbase = W + (size_t)m * D_DIM * DOUT;

    // Xp (512x16 per wave), paired d-chunks for WMMA ILP
    #pragma unroll 1
    for (int dp = 0; dp < 16; ++dp) {
      const int dA = dp * 32, dB = dp * 32 + 16;
      v8f a0 = {}, a1 = {};
      #pragma unroll
      for (int t = 0; t < 16; ++t) {
        const int oo = t * 4 + 2 * half;
        v2f fA = *(const v2f*)(wbase + (size_t)(dA + ln) * DOUT + oo); // A[d][o] = W_m
        v2f fB = *(const v2f*)(wbase + (size_t)(dB + ln) * DOUT + oo);
        a0 = wmma4(fA, bf[t], a0);
        a1 = wmma4(fB, bf[t], a1);
      }
      #pragma unroll
      for (int s = 0; s < 2; ++s) {
        const v8f& a = s ? a1 : a0;
        const int d = (s ? dB : dA) + half * 8;
        v4f lo = {a[0], a[1], a[2], a[3]};
        v4f hi = {a[4], a[5], a[6], a[7]};
        float* gp = o_xp + ((size_t)bcol * M_DIM + m) * D_DIM + d;
        *(v4f*)gp = lo;
        *(v4f*)(gp + 4) = hi;
        float* sp = xt + (wb + ln) * XT_PITCH + d;   // Xp-sum accumulation (wave-private)
        v4f s0 = *(v4f*)sp, s1 = *(v4f*)(sp + 4);
        s0 += lo; s1 += hi;
        *(v4f*)sp = s0;
        *(v4f*)(sp + 4) = s1;
      }
    }

    // scores: g[k] = ||c_k||^2 - 2 * (a . cproj[m,k,:]); running argmin per column
    float minv = 3.0e38f;
    int mk = 0;
    #pragma unroll 1
    for (int g = 0; g < 4; ++g) {
      v8f sa[4] = {};
      #pragma unroll
      for (int t = 0; t < 16; ++t) {
        const int oo = t * 4 + 2 * half;
        #pragma unroll
        for (int q = 0; q < 4; ++q) {
          v2f fa = *(const v2f*)(cproj + ((size_t)m * K_CNT + (g * 4 + q) * 16 + ln) * DOUT + oo);
          sa[q] = wmma4(fa, bf[t], sa[q]);
        }
      }
      #pragma unroll
      for (int q = 0; q < 4; ++q) {
        const int kb = (g * 4 + q) * 16 + half * 8;  // k ascending within each lane
        #pragma unroll
        for (int r = 0; r < 8; ++r) {
          float val = lcn[kb + r] - 2.0f * sa[q][r];
          if (val < minv) { minv = val; mk = kb + r; }   // strict <: first index on ties
        }
      }
    }
    // combine lane halves (lanes L and L+16 hold complementary k rows of column b)
    float ov = __shfl_xor(minv, 16, 32);
    int   okk = __shfl_xor(mk, 16, 32);
    if (ov < minv || (ov == minv && okk < mk)) { minv = ov; mk = okk; }
    if (lane < 16) {
      lbl[m * 64 + wb + lane] = mk;
      o_lbl[(size_t)(b0 + wb + lane) * M_DIM + m] = (float)mk;
    }
  }

  // ---- write Xp_sum (from LDS accumulator) ----
  #pragma unroll 4
  for (int dc = 0; dc < 32; ++dc) {
    const int d = dc * 16 + half * 8;
    float* sp = xt + (wb + ln) * XT_PITCH + d;
    v4f lo = *(v4f*)sp, hi = *(v4f*)(sp + 4);
    float* gp = o_xps + (size_t)bcol * D_DIM + d;
    *(v4f*)gp = lo;
    *(v4f*)(gp + 4) = hi;
  }

  // ---- Xr = center[label], Xr_sum, fully coalesced float4 per lane ----
  #pragma unroll 1
  for (int c = 0; c < 16; ++c) {
    const int bc = b0 + wb + c;
    v4f s0 = {}, s1 = {}, s2 = {}, s3 = {};
    #pragma unroll 1
    for (int m = 0; m < M_DIM; ++m) {
      const int kk = lbl[m * 64 + wb + c];
      const float* cr = center + (size_t)kk * D_DIM + lane * 4;
      float* xr = o_xr + ((size_t)bc * M_DIM + m) * D_DIM + lane * 4;
      v4f v0 = *(const v4f*)(cr + 0);
      v4f v1 = *(const v4f*)(cr + 128);
      v4f v2 = *(const v4f*)(cr + 256);
      v4f v3 = *(const v4f*)(cr + 384);
      s0 += v0; s1 += v1; s2 += v2; s3 += v3;
      *(v4f*)(xr + 0)   = v0;
      *(v4f*)(xr + 128) = v1;
      *(v4f*)(xr + 256) = v2;
      *(v4f*)(xr + 384) = v3;
    }
    float* xs = o_xrs + (size_t)bc * D_DIM + lane * 4;
    *(v4f*)(xs + 0)   = s0;
    *(v4f*)(xs + 128) = s1;
    *(v4f*)(xs + 256) = s2;
    *(v4f*)(xs + 384) = s3;
  }
}

extern "C" void kernel_launch(void* const* d_in, const int* in_sizes, int n_in,
                              void* d_out, int out_size, void* d_ws, size_t ws_size,
                              hipStream_t stream) {
  (void)in_sizes; (void)n_in; (void)out_size; (void)ws_size;
  const float* X = (const float*)d_in[0];   // [16384,512]
  const float* W = (const float*)d_in[1];   // [8,512,64]
  const float* C = (const float*)d_in[2];   // [256,512]

  float* out   = (float*)d_out;
  float* o_xr  = out;
  float* o_xp  = o_xr  + (size_t)B_TOT * M_DIM * D_DIM;
  float* o_xrs = o_xp  + (size_t)B_TOT * M_DIM * D_DIM;
  float* o_xps = o_xrs + (size_t)B_TOT * D_DIM;
  float* o_ctr = o_xps + (size_t)B_TOT * D_DIM;
  float* o_lbl = o_ctr + (size_t)K_CNT * D_DIM;

  float* cproj = (float*)d_ws;              // 131072 floats
  float* cnorm = cproj + M_DIM * K_CNT * DOUT; // 256 floats

  prep_cproj<<<M_DIM * K_CNT, 64, 0, stream>>>(W, C, cproj);
  prep_cnorm<<<K_CNT, 128, 0, stream>>>(C, cnorm, o_ctr);

  const size_t smem = (size_t)(64 * XT_PITCH + M_DIM * 64 * AT_PITCH) * sizeof(float)
                    + (size_t)M_DIM * 64 * sizeof(int) + K_CNT * sizeof(float); // 274,432 B
  projkmeans_main<<<B_TOT / 64, 128, smem, stream>>>(X, W, cproj, cnorm, C,
                                                     o_xr, o_xp, o_xrs, o_xps, o_lbl);
}